// InferenceXtt_28767690949073
// MI455X (gfx1250) — compile-verified
//
#include <hip/hip_runtime.h>
#include <hip/hip_bf16.h>

#define MQ 32
#define NG 256
#define DD 128

typedef int v4i __attribute__((ext_vector_type(4)));

// ---------- wave32 helpers ----------
__device__ __forceinline__ float wave_sum(float v) {
#pragma unroll
  for (int off = 16; off > 0; off >>= 1)
    v += __shfl_xor(v, off, 32);
  return v;
}

// CDNA5 async global -> LDS copy (128-bit per lane), tracked by ASYNCcnt.
__device__ __forceinline__ void async_b128(const float* g, float* l) {
#if defined(__has_builtin) && __has_builtin(__builtin_amdgcn_global_load_async_to_lds_b128)
  __builtin_amdgcn_global_load_async_to_lds_b128(
      (__attribute__((address_space(1))) v4i*)(g),
      (__attribute__((address_space(3))) v4i*)(l),
      /*imm offset*/ 0, /*cpol*/ 0);
#else
  // GV mode: vdst = LDS byte address, vaddr = 64-bit global address, saddr = off
  unsigned lds_addr = (unsigned)(size_t)(void*)l;  // low 32 bits of generic ptr = LDS offset
  asm volatile("global_load_async_to_lds_b128 %0, %1, off"
               :: "v"(lds_addr), "v"(g)
               : "memory");
#endif
}

__device__ __forceinline__ void wait_async0() {
#if defined(__has_builtin) && __has_builtin(__builtin_amdgcn_s_wait_asynccnt)
  __builtin_amdgcn_s_wait_asynccnt(0);
#else
  asm volatile("s_wait_asynccnt 0" ::: "memory");
#endif
}

// ---------- kernel ----------
// grid = 256 blocks (32 queries x 8 j-blocks), block = 256 threads = 8 wave32s.
// Each wave computes 4 (i,j) pairs. Mp lives in LDS (64 KB), loaded via async DMA.
__global__ __launch_bounds__(256) void jaccard_pen_kernel(
    const float* __restrict__ qf, const float* __restrict__ x,
    const float* __restrict__ Mp, const float* __restrict__ ze,
    float* __restrict__ out) {
  __shared__ float4 shMp4[DD * (DD / 4)];  // 128x128 f32 = 64 KB

  const int tid  = threadIdx.x;
  const int lane = tid & 31;
  const int wave = tid >> 5;

  // Stage Mp into LDS: 64 KB / (256 threads * 16 B) = 16 async b128 per thread.
  {
    const float* gbase = Mp + tid * 4;
    float*       lbase = reinterpret_cast<float*>(shMp4) + tid * 4;
#pragma unroll
    for (int k = 0; k < 16; ++k)
      async_b128(gbase + k * 1024, lbase + k * 1024);
  }

  const int   bid = blockIdx.x;
  const int   i   = bid >> 3;          // query index 0..31
  const int   jb  = (bid & 7) * 32;    // gallery block base
  const float zev = ze[0];

  // This wave's slice of the query row: d = 4*lane .. 4*lane+3
  const float4 q4 = reinterpret_cast<const float4*>(qf)[i * (DD / 4) + lane];

  wait_async0();     // own async copies done
  __syncthreads();   // all waves' slices of Mp visible

  for (int p = 0; p < 4; ++p) {
    const int    j  = jb + p * 8 + wave;
    const float4 x4 = reinterpret_cast<const float4*>(x)[j * (DD / 4) + lane];

    float4 u, v;
    u.x = fmaxf(q4.x, x4.x); v.x = fminf(q4.x, x4.x);
    u.y = fmaxf(q4.y, x4.y); v.y = fminf(q4.y, x4.y);
    u.z = fmaxf(q4.z, x4.z); v.z = fminf(q4.z, x4.z);
    u.w = fmaxf(q4.w, x4.w); v.w = fminf(q4.w, x4.w);

    const float su = wave_sum(u.x + u.y + u.z + u.w);
    const float sv = wave_sum(v.x + v.y + v.z + v.w);

    // pen = sum_{a,b} max(zev, u_a*u_b - Mp[a][b]); this lane owns b = 4*lane+c.
    float acc0 = 0.f, acc1 = 0.f, acc2 = 0.f, acc3 = 0.f;
#pragma unroll 4
    for (int ah = 0; ah < 32; ++ah) {
      float ua[4];
      ua[0] = __shfl(u.x, ah, 32);   // u_a for a = 4*ah + 0..3 (uniform src lane)
      ua[1] = __shfl(u.y, ah, 32);
      ua[2] = __shfl(u.z, ah, 32);
      ua[3] = __shfl(u.w, ah, 32);
#pragma unroll
      for (int c = 0; c < 4; ++c) {
        const float4 m = shMp4[(ah * 4 + c) * (DD / 4) + lane];  // ds_load_b128
        float t;
        t = fmaf(ua[c], u.x, -m.x); acc0 += fmaxf(t, zev);
        t = fmaf(ua[c], u.y, -m.y); acc1 += fmaxf(t, zev);
        t = fmaf(ua[c], u.z, -m.z); acc2 += fmaxf(t, zev);
        t = fmaf(ua[c], u.w, -m.w); acc3 += fmaxf(t, zev);
      }
    }
    const float pen = wave_sum(acc0 + acc1 + acc2 + acc3);

    if (lane == 0)
      out[i * NG + j] = sv / su - 0.001f * pen;
  }
}

// ---------- launch ----------
extern "C" void kernel_launch(void* const* d_in, const int* in_sizes, int n_in,
                              void* d_out, int out_size, void* d_ws, size_t ws_size,
                              hipStream_t stream) {
  const float* qf = (const float*)d_in[0];  // (32, 128)
  const float* x  = (const float*)d_in[1];  // (256, 128)
  const float* Mp = (const float*)d_in[2];  // (128, 128)
  const float* ze = (const float*)d_in[3];  // scalar
  float*       out = (float*)d_out;         // (32, 256)

  dim3 grid(256), block(256);
  hipLaunchKernelGGL(jaccard_pen_kernel, grid, block, 0, stream,
                     qf, x, Mp, ze, out);
}